// PoseDecoderSparse_10127532884218
// MI455X (gfx1250) — compile-verified
//
#include <hip/hip_runtime.h>
#include <hip/hip_bf16.h>

typedef float v2f __attribute__((ext_vector_type(2)));
typedef float v8f __attribute__((ext_vector_type(8)));

#define NPTS 4096
#define BLK 256
#define NWAVE (BLK / 32)

// ---------- small device linear algebra helpers ----------

__device__ inline void inv3(const float* K, float* Ki) {
  float a = K[0], b = K[1], c = K[2];
  float d = K[3], e = K[4], f = K[5];
  float g = K[6], h = K[7], i = K[8];
  float det = a * (e * i - f * h) - b * (d * i - f * g) + c * (d * h - e * g);
  float inv = 1.0f / det;
  Ki[0] = (e * i - f * h) * inv;
  Ki[1] = (c * h - b * i) * inv;
  Ki[2] = (b * f - c * e) * inv;
  Ki[3] = (f * g - d * i) * inv;
  Ki[4] = (a * i - c * g) * inv;
  Ki[5] = (c * d - a * f) * inv;
  Ki[6] = (d * h - e * g) * inv;
  Ki[7] = (b * g - a * h) * inv;
  Ki[8] = (a * e - b * d) * inv;
}

__device__ inline float det3f(const float* M) {
  return M[0] * (M[4] * M[8] - M[5] * M[7])
       - M[1] * (M[3] * M[8] - M[5] * M[6])
       + M[2] * (M[3] * M[7] - M[4] * M[6]);
}

// Cyclic Jacobi eigensolver for symmetric n x n (n <= 9).
// On exit: eigenvalues on diag(A), eigenvectors in columns of V.
__device__ void jacobi_sym(float* A, float* V, int n, int sweeps) {
  for (int i = 0; i < n; i++)
    for (int j = 0; j < n; j++) V[i * n + j] = (i == j) ? 1.0f : 0.0f;
  for (int s = 0; s < sweeps; s++) {
    for (int p = 0; p < n - 1; p++) {
      for (int q = p + 1; q < n; q++) {
        float apq = A[p * n + q];
        if (fabsf(apq) < 1e-30f) continue;
        float app = A[p * n + p], aqq = A[q * n + q];
        float theta = 0.5f * (aqq - app) / apq;
        float t = copysignf(1.0f, theta) / (fabsf(theta) + sqrtf(theta * theta + 1.0f));
        float c = 1.0f / sqrtf(t * t + 1.0f);
        float sn = t * c;
        for (int k = 0; k < n; k++) {
          float akp = A[k * n + p], akq = A[k * n + q];
          A[k * n + p] = c * akp - sn * akq;
          A[k * n + q] = sn * akp + c * akq;
        }
        for (int k = 0; k < n; k++) {
          float apk = A[p * n + k], aqk = A[q * n + k];
          A[p * n + k] = c * apk - sn * aqk;
          A[q * n + k] = sn * apk + c * aqk;
        }
        for (int k = 0; k < n; k++) {
          float vkp = V[k * n + p], vkq = V[k * n + q];
          V[k * n + p] = c * vkp - sn * vkq;
          V[k * n + q] = sn * vkp + c * vkq;
        }
      }
    }
  }
}

// ---------- fused pose kernel: one workgroup per batch ----------

__global__ __launch_bounds__(BLK) void pose8pt_kernel(
    const float* __restrict__ kpts0, const float* __restrict__ kpts1,
    const float* __restrict__ conf, const float* __restrict__ t_scale,
    const float* __restrict__ Kmat, float* __restrict__ out) {
  __shared__ float sKinv0[9];
  __shared__ float sKinv1[9];
  __shared__ float sMpart[NWAVE][256];  // per-wave 16x16 accumulator dump
  __shared__ float sMfull[256];         // reduced 16x16 Gram (9x9 valid)
  __shared__ float sRc[4][9];
  __shared__ float sTc[4][3];
  __shared__ float sScore[4][BLK];

  const int b = blockIdx.x;
  const int tid = threadIdx.x;
  const int lane = tid & 31;
  const int wave = tid >> 5;

  if (tid == 0) {
    float Kl[9];
    for (int i = 0; i < 9; i++) Kl[i] = Kmat[(size_t)b * 18 + i];
    inv3(Kl, sKinv0);
    for (int i = 0; i < 9; i++) Kl[i] = Kmat[(size_t)b * 18 + 9 + i];
    inv3(Kl, sKinv1);
  }
  __syncthreads();

  float i0[9], i1[9];
  for (int i = 0; i < 9; i++) { i0[i] = sKinv0[i]; i1[i] = sKinv1[i]; }

  const float* kp0 = kpts0 + (size_t)b * NPTS * 2;
  const float* kp1 = kpts1 + (size_t)b * NPTS * 2;
  const float* cw  = conf  + (size_t)b * NPTS;

  // ---- Phase 1: M = X^T diag(w) X via chained V_WMMA_F32_16X16X4_F32 ----
  // A (16x4): A[i][k] = w_k * X_k[i]   (rows i<9 valid, padded with 0)
  // B (4x16): B[k][j] = X_k[j]
  // Per-lane layout (ISA 32-bit 16x4): lanes 0-15 hold K=0 (vgpr0) / K=1 (vgpr1),
  // lanes 16-31 hold K=2 / K=3; element index = lane % 16.
  const int half = lane >> 4;
  const int e = lane & 15;
  const int er = e / 3, ec = e - er * 3;  // X element e = x1[er]*x0[ec]
  const bool ev = (e < 9);

  // Branch-free one-hot selection weights (padding rows e>=9 folded into sb*):
  // keeps control flow uniform so EXEC is all-ones at every WMMA.
  const float sa0 = (ec == 0) ? 1.0f : 0.0f;
  const float sa1 = (ec == 1) ? 1.0f : 0.0f;
  const float sa2 = (ec == 2) ? 1.0f : 0.0f;
  const float sb0 = (ev && er == 0) ? 1.0f : 0.0f;
  const float sb1 = (ev && er == 1) ? 1.0f : 0.0f;
  const float sb2 = (ev && er == 2) ? 1.0f : 0.0f;

  // Vectorized pair loads: p0 is always even; one float4 covers both points.
  const float4* kp0v = reinterpret_cast<const float4*>(kp0);
  const float4* kp1v = reinterpret_cast<const float4*>(kp1);
  const float2* cwv  = reinterpret_cast<const float2*>(cw);

  v8f acc;
  for (int i = 0; i < 8; i++) acc[i] = 0.0f;

  const int pairbase = (wave * (NPTS / NWAVE)) >> 1;  // in point-pair units
  for (int c4 = 0; c4 < (NPTS / NWAVE) / 4; ++c4) {
    const int pp = pairbase + c4 * 2 + half;  // pair index: points 2pp, 2pp+1
    const float4 q0 = kp0v[pp];               // (u_a, v_a, u_b, v_b) in image 0
    const float4 q1 = kp1v[pp];               // (u_a, v_a, u_b, v_b) in image 1
    const float2 wv = cwv[pp];                // (w_a, w_b)

    // point a
    float x00 = i0[0] * q0.x + i0[1] * q0.y + i0[2];
    float x01 = i0[3] * q0.x + i0[4] * q0.y + i0[5];
    float x02 = i0[6] * q0.x + i0[7] * q0.y + i0[8];
    float x10 = i1[0] * q1.x + i1[1] * q1.y + i1[2];
    float x11 = i1[3] * q1.x + i1[4] * q1.y + i1[5];
    float x12 = i1[6] * q1.x + i1[7] * q1.y + i1[8];
    float xa = (sa0 * x00 + sa1 * x01 + sa2 * x02) *
               (sb0 * x10 + sb1 * x11 + sb2 * x12);
    // point b
    x00 = i0[0] * q0.z + i0[1] * q0.w + i0[2];
    x01 = i0[3] * q0.z + i0[4] * q0.w + i0[5];
    x02 = i0[6] * q0.z + i0[7] * q0.w + i0[8];
    x10 = i1[0] * q1.z + i1[1] * q1.w + i1[2];
    x11 = i1[3] * q1.z + i1[4] * q1.w + i1[5];
    x12 = i1[6] * q1.z + i1[7] * q1.w + i1[8];
    float xb = (sa0 * x00 + sa1 * x01 + sa2 * x02) *
               (sb0 * x10 + sb1 * x11 + sb2 * x12);

    v2f Av, Bv;
    Av[0] = wv.x * xa; Av[1] = wv.y * xb;
    Bv[0] = xa;        Bv[1] = xb;
    acc = __builtin_amdgcn_wmma_f32_16x16x4_f32(
        /*neg_a=*/false, Av, /*neg_b=*/false, Bv,
        /*c_mod=*/(short)0, acc, /*reuse_a=*/false, /*reuse_b=*/false);
  }

  for (int i = 0; i < 8; i++) sMpart[wave][lane * 8 + i] = acc[i];
  __syncthreads();

  // Deterministic cross-wave reduction of the 16x16 accumulator.
  {
    float s = 0.0f;
    for (int w = 0; w < NWAVE; w++) s += sMpart[w][tid];
    const int l  = tid >> 3, v = tid & 7;
    const int h2 = l >> 4, ee = l & 15;
    // C/D layout: vgpr v, lanes 0-15 -> row v; lanes 16-31 -> row v+8; col = lane%16
    sMfull[(v + 8 * h2) * 16 + ee] = s;
  }
  __syncthreads();

  // ---- Phase 2 (thread 0): eigensolve, 3x3 SVD, pose candidates ----
  if (tid == 0) {
    float A9[81], V9[81];
    for (int i = 0; i < 9; i++)
      for (int j = 0; j < 9; j++) A9[i * 9 + j] = sMfull[i * 16 + j];
    jacobi_sym(A9, V9, 9, 12);
    int mi = 0; float mv = A9[0];
    for (int i = 1; i < 9; i++) {
      float d = A9[i * 9 + i];
      if (d < mv) { mv = d; mi = i; }
    }
    float E[9];
    for (int i = 0; i < 9; i++) E[i] = V9[i * 9 + mi];  // E[r][c]=v[r*3+c]

    // SVD of E via eig of G = E^T E
    float G[9];
    for (int r = 0; r < 3; r++)
      for (int c = 0; c < 3; c++) {
        float s = 0.0f;
        for (int k = 0; k < 3; k++) s += E[k * 3 + r] * E[k * 3 + c];
        G[r * 3 + c] = s;
      }
    float Vg[9];
    jacobi_sym(G, Vg, 3, 10);
    float lam[3] = { G[0], G[4], G[8] };
    int idx[3] = { 0, 1, 2 };
    if (lam[idx[0]] < lam[idx[1]]) { int t2 = idx[0]; idx[0] = idx[1]; idx[1] = t2; }
    if (lam[idx[0]] < lam[idx[2]]) { int t2 = idx[0]; idx[0] = idx[2]; idx[2] = t2; }
    if (lam[idx[1]] < lam[idx[2]]) { int t2 = idx[1]; idx[1] = idx[2]; idx[2] = t2; }
    float V3[9];
    for (int c = 0; c < 3; c++)
      for (int r = 0; r < 3; r++) V3[r * 3 + c] = Vg[r * 3 + idx[c]];

    float U[9];
    for (int c = 0; c < 2; c++) {
      float uc[3];
      for (int r = 0; r < 3; r++)
        uc[r] = E[r * 3 + 0] * V3[0 * 3 + c] + E[r * 3 + 1] * V3[1 * 3 + c] +
                E[r * 3 + 2] * V3[2 * 3 + c];
      if (c == 1) {  // re-orthogonalize against u0
        float d = uc[0] * U[0] + uc[1] * U[3] + uc[2] * U[6];
        uc[0] -= d * U[0]; uc[1] -= d * U[3]; uc[2] -= d * U[6];
      }
      float nr = sqrtf(uc[0] * uc[0] + uc[1] * uc[1] + uc[2] * uc[2]) + 1e-20f;
      U[0 * 3 + c] = uc[0] / nr; U[1 * 3 + c] = uc[1] / nr; U[2 * 3 + c] = uc[2] / nr;
    }
    // u2 = u0 x u1 (null-space left singular vector)
    U[0 * 3 + 2] = U[1 * 3 + 0] * U[2 * 3 + 1] - U[2 * 3 + 0] * U[1 * 3 + 1];
    U[1 * 3 + 2] = U[2 * 3 + 0] * U[0 * 3 + 1] - U[0 * 3 + 0] * U[2 * 3 + 1];
    U[2 * 3 + 2] = U[0 * 3 + 0] * U[1 * 3 + 1] - U[1 * 3 + 0] * U[0 * 3 + 1];

    // R1 = U W V^T, R2 = U W^T V^T, W = [[0,-1,0],[1,0,0],[0,0,1]]
    float UW[9], UWt[9];
    for (int r = 0; r < 3; r++) {
      UW[r * 3 + 0]  =  U[r * 3 + 1];
      UW[r * 3 + 1]  = -U[r * 3 + 0];
      UW[r * 3 + 2]  =  U[r * 3 + 2];
      UWt[r * 3 + 0] = -U[r * 3 + 1];
      UWt[r * 3 + 1] =  U[r * 3 + 0];
      UWt[r * 3 + 2] =  U[r * 3 + 2];
    }
    float R1[9], R2[9];
    for (int r = 0; r < 3; r++)
      for (int c = 0; c < 3; c++) {
        float s1 = 0.0f, s2 = 0.0f;
        for (int k = 0; k < 3; k++) {
          s1 += UW[r * 3 + k]  * V3[c * 3 + k];
          s2 += UWt[r * 3 + k] * V3[c * 3 + k];
        }
        R1[r * 3 + c] = s1;
        R2[r * 3 + c] = s2;
      }
    float sg1 = (det3f(R1) < 0.0f) ? -1.0f : 1.0f;
    float sg2 = (det3f(R2) < 0.0f) ? -1.0f : 1.0f;
    for (int i = 0; i < 9; i++) { R1[i] *= sg1; R2[i] *= sg2; }
    float t[3] = { U[2], U[5], U[8] };

    for (int i = 0; i < 9; i++) {
      sRc[0][i] = R1[i]; sRc[1][i] = R1[i];
      sRc[2][i] = R2[i]; sRc[3][i] = R2[i];
    }
    for (int i = 0; i < 3; i++) {
      sTc[0][i] = t[i];  sTc[1][i] = -t[i];
      sTc[2][i] = t[i];  sTc[3][i] = -t[i];
    }
  }
  __syncthreads();

  // ---- Phase 3: cheirality scoring over all N points ----
  float sc[4] = { 0.0f, 0.0f, 0.0f, 0.0f };
  const float2* kp0p = reinterpret_cast<const float2*>(kp0);
  const float2* kp1p = reinterpret_cast<const float2*>(kp1);
  for (int p = tid; p < NPTS; p += BLK) {
    const float2 a0p = kp0p[p];
    const float2 a1p = kp1p[p];
    float x00 = i0[0] * a0p.x + i0[1] * a0p.y + i0[2];
    float x01 = i0[3] * a0p.x + i0[4] * a0p.y + i0[5];
    float x02 = i0[6] * a0p.x + i0[7] * a0p.y + i0[8];
    float x10 = i1[0] * a1p.x + i1[1] * a1p.y + i1[2];
    float x11 = i1[3] * a1p.x + i1[4] * a1p.y + i1[5];
    float x12 = i1[6] * a1p.x + i1[7] * a1p.y + i1[8];
    float w = cw[p];
    float bb = x10 * x10 + x11 * x11 + x12 * x12;
    for (int c = 0; c < 4; c++) {
      const float* R  = sRc[c];
      const float* tc = sTc[c];
      float a0 = R[0] * x00 + R[1] * x01 + R[2] * x02;
      float a1 = R[3] * x00 + R[4] * x01 + R[5] * x02;
      float a2 = R[6] * x00 + R[7] * x01 + R[8] * x02;
      float aa = a0 * a0 + a1 * a1 + a2 * a2;
      float ab = a0 * x10 + a1 * x11 + a2 * x12;
      float at = a0 * tc[0] + a1 * tc[1] + a2 * tc[2];
      float bt = x10 * tc[0] + x11 * tc[1] + x12 * tc[2];
      float det = aa * bb - ab * ab + 1e-9f;
      float d0 = (-at * bb + ab * bt) / det;
      float d1 = (aa * bt - ab * at) / det;
      sc[c] += (d0 > 0.0f && d1 > 0.0f) ? w : 0.0f;
    }
  }
  for (int c = 0; c < 4; c++) sScore[c][tid] = sc[c];
  __syncthreads();
  for (int s = BLK / 2; s > 0; s >>= 1) {  // fixed-order tree: deterministic
    if (tid < s)
      for (int c = 0; c < 4; c++) sScore[c][tid] += sScore[c][tid + s];
    __syncthreads();
  }

  // ---- Phase 4 (thread 0): select best, assemble T01 and T10 ----
  if (tid == 0) {
    int best = 0; float bv = sScore[0][0];
    for (int c = 1; c < 4; c++)
      if (sScore[c][0] > bv) { bv = sScore[c][0]; best = c; }
    const float ts = t_scale[b * 2 + 0];
    const float* R  = sRc[best];
    const float* tc = sTc[best];
    float t0 = tc[0] * ts, t1 = tc[1] * ts, t2 = tc[2] * ts;
    float* o = out + (size_t)b * 32;
    // T01 = [R | t_sc ; 0 0 0 1]
    o[0]  = R[0]; o[1]  = R[1]; o[2]  = R[2]; o[3]  = t0;
    o[4]  = R[3]; o[5]  = R[4]; o[6]  = R[5]; o[7]  = t1;
    o[8]  = R[6]; o[9]  = R[7]; o[10] = R[8]; o[11] = t2;
    o[12] = 0.0f; o[13] = 0.0f; o[14] = 0.0f; o[15] = 1.0f;
    // T10 = [R^T | -R^T t_sc ; 0 0 0 1]
    float ti0 = -(R[0] * t0 + R[3] * t1 + R[6] * t2);
    float ti1 = -(R[1] * t0 + R[4] * t1 + R[7] * t2);
    float ti2 = -(R[2] * t0 + R[5] * t1 + R[8] * t2);
    o[16] = R[0]; o[17] = R[3]; o[18] = R[6]; o[19] = ti0;
    o[20] = R[1]; o[21] = R[4]; o[22] = R[7]; o[23] = ti1;
    o[24] = R[2]; o[25] = R[5]; o[26] = R[8]; o[27] = ti2;
    o[28] = 0.0f; o[29] = 0.0f; o[30] = 0.0f; o[31] = 1.0f;
  }
}

extern "C" void kernel_launch(void* const* d_in, const int* in_sizes, int n_in,
                              void* d_out, int out_size, void* d_ws, size_t ws_size,
                              hipStream_t stream) {
  const float* kpts0   = (const float*)d_in[0];  // (B, N, 2)
  const float* kpts1   = (const float*)d_in[1];  // (B, N, 2)
  const float* conf    = (const float*)d_in[2];  // (B, N, 1)
  const float* t_scale = (const float*)d_in[3];  // (B, 2)
  const float* K       = (const float*)d_in[4];  // (B, 2, 3, 3)
  // d_in[5] = gt_pose_0to1: unused by the reference output
  float* out = (float*)d_out;                    // (B, 2, 4, 4)

  const int B = in_sizes[3] / 2;
  pose8pt_kernel<<<B, BLK, 0, stream>>>(kpts0, kpts1, conf, t_scale, K, out);
}